// SimpleGCN_49916109914272
// MI455X (gfx1250) — compile-verified
//
#include <hip/hip_runtime.h>
#include <math.h>

#define WAVE 32
#define NEG_SLOPE 0.2f
#define HEADS 4            // constant across all three GAT layers
#define KC 64              // GEMM K-chunk staged in LDS (all K are multiples of 64)
#define BS_STRIDE 48       // 32 B-columns + 16 pad words -> conflict-free ds_load

typedef float v2f __attribute__((ext_vector_type(2)));
typedef float v8f __attribute__((ext_vector_type(8)));

// ---------- monotone float<->uint encoding for atomic max on floats ----------
__device__ __forceinline__ unsigned fenc(float f) {
  unsigned u = __float_as_uint(f);
  return (u & 0x80000000u) ? ~u : (u | 0x80000000u);
}
__device__ __forceinline__ float fdec(unsigned u) {
  u = (u & 0x80000000u) ? (u & 0x7FFFFFFFu) : ~u;
  return __uint_as_float(u);
}

// ---------- WMMA fp32 GEMM: C[M,N] = A[M,K] @ B[K,N] ----------
// Block = 4 waves -> 64(M) x 32(N) tile. B chunk [KC x 32] staged in LDS and
// shared by all 4 waves; each wave computes two 16x16 N-tiles reusing its A
// fragment across both WMMAs. Requires M%16==0, N%32==0, K%KC==0 (all hold).
__global__ __launch_bounds__(128)
void gemm_wmma_f32(const float* __restrict__ A, const float* __restrict__ B,
                   float* __restrict__ C, int M, int N, int K, int nBlocksM) {
  __shared__ float Bs[KC * BS_STRIDE];   // 12 KB
  const int tid  = (int)threadIdx.x;
  const int wv   = tid / WAVE;           // wave in block (0..3)
  const int lane = tid & (WAVE - 1);
  const int half = lane >> 4;            // 0: K {0,1}, 1: K {2,3}
  const int l16  = lane & 15;

  const int mb = (int)blockIdx.x % nBlocksM;
  const int nb = (int)blockIdx.x / nBlocksM;
  const int mt = mb * 4 + wv;            // this wave's M-tile
  const int mTiles = M >> 4;
  const bool valid = (mt < mTiles);      // wave-uniform -> EXEC stays all-ones
  const int colBase = nb * 32;

  const long aRow = (long)(mt * 16 + l16) * K;  // A: lane holds row M=l16 (both halves)
  v8f acc0 = {0.f,0.f,0.f,0.f,0.f,0.f,0.f,0.f};
  v8f acc1 = {0.f,0.f,0.f,0.f,0.f,0.f,0.f,0.f};

  for (int k0 = 0; k0 < K; k0 += KC) {
    // stage B[k0..k0+KC) x 32 cols into LDS (coalesced, conflict-free writes)
    for (int i = tid; i < KC * 32; i += 128) {
      int k = i >> 5, c = i & 31;
      Bs[k * BS_STRIDE + c] = B[(long)(k0 + k) * N + colBase + c];
    }
    __syncthreads();
    if (valid) {
      if (k0 + KC < K)  // prefetch next A chunk (global_prefetch_b8)
        __builtin_prefetch(&A[aRow + k0 + KC], 0, 1);
      for (int kk = 0; kk < KC; kk += 4) {
        const int ka = kk + half * 2;
        v2f a, b0, b1;
        a.x  = A[aRow + k0 + ka];                 // A vgpr0 = K {0|2}
        a.y  = A[aRow + k0 + ka + 1];             // A vgpr1 = K {1|3}
        b0.x = Bs[ka * BS_STRIDE + l16];          // B rows striped across lanes/halves
        b0.y = Bs[(ka + 1) * BS_STRIDE + l16];
        b1.x = Bs[ka * BS_STRIDE + 16 + l16];
        b1.y = Bs[(ka + 1) * BS_STRIDE + 16 + l16];
        acc0 = __builtin_amdgcn_wmma_f32_16x16x4_f32(
            false, a, false, b0, (short)0, acc0, false, false);
        acc1 = __builtin_amdgcn_wmma_f32_16x16x4_f32(
            false, a, false, b1, (short)0, acc1, false, false);
      }
    }
    __syncthreads();
  }
  if (valid) {
    const long cRow0 = (long)(mt * 16 + half * 8);  // C/D: vgpr i -> M = i (+8 hi half)
#pragma unroll
    for (int i = 0; i < 8; ++i) {
      C[(cRow0 + i) * N + colBase + l16]      = acc0[i];
      C[(cRow0 + i) * N + colBase + 16 + l16] = acc1[i];
    }
  }
}

// ---------- per-node per-head attention score contributions ----------
__global__ void node_scores(const float* __restrict__ h, const float* __restrict__ asr,
                            const float* __restrict__ adr, float* __restrict__ es,
                            float* __restrict__ ed, int N, int C) {
  long idx = (long)blockIdx.x * blockDim.x + threadIdx.x;
  if (idx >= (long)N * HEADS) return;
  int n = (int)(idx >> 2), hh = (int)(idx & 3);
  const float* hp = h + ((long)n * HEADS + hh) * C;
  float s = 0.f, d = 0.f;
  for (int c = 0; c < C; ++c) {
    float v = hp[c];
    s += v * asr[hh * C + c];
    d += v * adr[hh * C + c];
  }
  es[idx] = s;
  ed[idx] = d;
}

__device__ __forceinline__ void edge_sd(const int* __restrict__ ei, int E, int eid,
                                        int& s, int& d) {
  if (eid < E) { s = ei[eid]; d = ei[E + eid]; }
  else         { s = eid - E; d = eid - E; }       // self-loops appended
}

__device__ __forceinline__ float lrelu(float x) {
  return x > 0.f ? x : NEG_SLOPE * x;
}

// ---------- pass A: segment max over incoming edges ----------
__global__ void edge_max_kernel(const int* __restrict__ ei, int E, long totEH,
                                const float* __restrict__ es, const float* __restrict__ ed,
                                unsigned* __restrict__ menc) {
  long idx = (long)blockIdx.x * blockDim.x + threadIdx.x;
  if (idx >= totEH) return;
  int eid = (int)(idx >> 2), hh = (int)(idx & 3);
  int s, d; edge_sd(ei, E, eid, s, d);
  float e = lrelu(es[(long)s * HEADS + hh] + ed[(long)d * HEADS + hh]);
  atomicMax(&menc[(long)d * HEADS + hh], fenc(e));
}

__global__ void finalize_max(const unsigned* __restrict__ menc, float* __restrict__ m, long n) {
  long idx = (long)blockIdx.x * blockDim.x + threadIdx.x;
  if (idx >= n) return;
  float v = fdec(menc[idx]);
  m[idx] = isfinite(v) ? v : 0.f;          // reference: where(isfinite(m), m, 0)
}

// ---------- pass B: exp-sum denominator ----------
__global__ void edge_expsum(const int* __restrict__ ei, int E, long totEH,
                            const float* __restrict__ es, const float* __restrict__ ed,
                            const float* __restrict__ m, float* __restrict__ den) {
  long idx = (long)blockIdx.x * blockDim.x + threadIdx.x;
  if (idx >= totEH) return;
  int eid = (int)(idx >> 2), hh = (int)(idx & 3);
  int s, d; edge_sd(ei, E, eid, s, d);
  float e = lrelu(es[(long)s * HEADS + hh] + ed[(long)d * HEADS + hh]);
  atomicAdd(&den[(long)d * HEADS + hh], expf(e - m[(long)d * HEADS + hh]));
}

// ---------- pass B2: per-(edge,head) alpha ----------
__global__ void edge_alpha(const int* __restrict__ ei, int E, long totEH,
                           const float* __restrict__ es, const float* __restrict__ ed,
                           const float* __restrict__ m, const float* __restrict__ den,
                           float* __restrict__ alpha) {
  long idx = (long)blockIdx.x * blockDim.x + threadIdx.x;
  if (idx >= totEH) return;
  int eid = (int)(idx >> 2), hh = (int)(idx & 3);
  int s, d; edge_sd(ei, E, eid, s, d);
  float e = lrelu(es[(long)s * HEADS + hh] + ed[(long)d * HEADS + hh]);
  float ex = expf(e - m[(long)d * HEADS + hh]);
  alpha[idx] = ex / (den[(long)d * HEADS + hh] + 1e-16f);
}

// ---------- pass C: weighted scatter-aggregate (float4 per thread) ----------
__global__ void edge_aggregate(const int* __restrict__ ei, int E, long total,
                               const float* __restrict__ alpha, const float* __restrict__ h,
                               float* __restrict__ out, int C, int cqshift, int concat) {
  long idx = (long)blockIdx.x * blockDim.x + threadIdx.x;
  if (idx >= total) return;
  const int cqmask = (1 << cqshift) - 1;       // C/4 is a power of two
  int c4  = (int)(idx & cqmask);
  long t  = idx >> cqshift;
  int hh  = (int)(t & 3);
  int eid = (int)(t >> 2);
  int s, d; edge_sd(ei, E, eid, s, d);
  float a = alpha[(long)eid * HEADS + hh];
  const float4 hv = *(const float4*)&h[((long)s * HEADS + hh) * C + c4 * 4];
  if (concat) {
    float* o = &out[((long)d * HEADS + hh) * C + c4 * 4];
    atomicAdd(o + 0, a * hv.x); atomicAdd(o + 1, a * hv.y);
    atomicAdd(o + 2, a * hv.z); atomicAdd(o + 3, a * hv.w);
  } else {
    const float sc = a * (1.0f / HEADS);
    float* o = &out[(long)d * C + c4 * 4];
    atomicAdd(o + 0, sc * hv.x); atomicAdd(o + 1, sc * hv.y);
    atomicAdd(o + 2, sc * hv.z); atomicAdd(o + 3, sc * hv.w);
  }
}

// ---------- bias + ELU, in place ----------
__global__ void bias_elu(float* __restrict__ x, const float* __restrict__ b, long n, int F) {
  long idx = (long)blockIdx.x * blockDim.x + threadIdx.x;
  if (idx >= n) return;
  float v = x[idx] + b[idx & (F - 1)];         // F is a power of two (64/256/128)
  x[idx] = v > 0.f ? v : expm1f(v);
}

// ---------- global mean pool + final linear ----------
__global__ void pool_sum(const float* __restrict__ h, const int* __restrict__ batch,
                         float* __restrict__ ps, int N, int F) {
  long idx = (long)blockIdx.x * blockDim.x + threadIdx.x;
  if (idx >= (long)N * F) return;
  int n = (int)(idx / F), c = (int)(idx % F);
  atomicAdd(&ps[(long)batch[n] * F + c], h[idx]);
}

__global__ void pool_cnt(const int* __restrict__ batch, float* __restrict__ cnt, int N) {
  long idx = (long)blockIdx.x * blockDim.x + threadIdx.x;
  if (idx >= N) return;
  atomicAdd(&cnt[batch[idx]], 1.0f);
}

__global__ void final_linear(const float* __restrict__ ps, const float* __restrict__ cnt,
                             const float* __restrict__ Wl, const float* __restrict__ bl,
                             float* __restrict__ out, int G, int F, int O) {
  long idx = (long)blockIdx.x * blockDim.x + threadIdx.x;
  if (idx >= (long)G * O) return;
  int g = (int)(idx / O), j = (int)(idx % O);
  float inv = 1.0f / fmaxf(cnt[g], 1.0f);
  float acc = bl[j];
  for (int c = 0; c < F; ++c) acc += (ps[(long)g * F + c] * inv) * Wl[c * O + j];
  out[idx] = acc;
}

// ---------------------------- host side ----------------------------
static inline int gblk(long n, int bs) { return (int)((n + bs - 1) / bs); }

static void run_gat_layer(const float* x, int N, int Fin, int C, int concat,
                          const float* W, const float* asr, const float* adr, const float* b,
                          const int* ei, int E,
                          float* h, float* out,
                          float* es, float* ed, unsigned* menc, float* m, float* den,
                          float* alpha, hipStream_t stream) {
  const int Fout = HEADS * C;
  const int ET = E + N;                 // self-loops appended
  const long NH = (long)N * HEADS;
  const long EH = (long)ET * HEADS;

  // h = x @ W   (WMMA fp32, LDS-staged B)
  {
    const int nBlocksM = (N / 16 + 3) / 4;
    const int nBlocksN = Fout / 32;
    gemm_wmma_f32<<<nBlocksM * nBlocksN, 128, 0, stream>>>(x, W, h, N, Fout, Fin, nBlocksM);
  }
  // per-node score contributions
  node_scores<<<gblk(NH, 256), 256, 0, stream>>>(h, asr, adr, es, ed, N, C);
  // segment max
  hipMemsetAsync(menc, 0, NH * sizeof(unsigned), stream);   // 0 < fenc(any finite)
  edge_max_kernel<<<gblk(EH, 256), 256, 0, stream>>>(ei, E, EH, es, ed, menc);
  finalize_max<<<gblk(NH, 256), 256, 0, stream>>>(menc, m, NH);
  // exp-sum
  hipMemsetAsync(den, 0, NH * sizeof(float), stream);
  edge_expsum<<<gblk(EH, 256), 256, 0, stream>>>(ei, E, EH, es, ed, m, den);
  // alpha
  edge_alpha<<<gblk(EH, 256), 256, 0, stream>>>(ei, E, EH, es, ed, m, den, alpha);
  // scatter-aggregate, 4 channels per thread
  const int Fagg = concat ? Fout : C;
  hipMemsetAsync(out, 0, (long)N * Fagg * sizeof(float), stream);
  const int cqshift = __builtin_ctz(C / 4);
  long total = (long)ET * HEADS * (C / 4);
  edge_aggregate<<<gblk(total, 256), 256, 0, stream>>>(ei, E, total, alpha, h, out,
                                                       C, cqshift, concat);
  // bias + ELU in place
  bias_elu<<<gblk((long)N * Fagg, 256), 256, 0, stream>>>(out, b, (long)N * Fagg, Fagg);
}

extern "C" void kernel_launch(void* const* d_in, const int* in_sizes, int n_in,
                              void* d_out, int out_size, void* d_ws, size_t ws_size,
                              hipStream_t stream) {
  const float* x    = (const float*)d_in[0];
  const int*   ei   = (const int*)  d_in[1];   // [2,E]: row0=src, row1=dst
  const int*   bat  = (const int*)  d_in[2];
  const float* W1   = (const float*)d_in[3];
  const float* as1  = (const float*)d_in[4];
  const float* ad1  = (const float*)d_in[5];
  const float* b1   = (const float*)d_in[6];
  const float* W2   = (const float*)d_in[7];
  const float* as2  = (const float*)d_in[8];
  const float* ad2  = (const float*)d_in[9];
  const float* b2   = (const float*)d_in[10];
  const float* W3   = (const float*)d_in[11];
  const float* as3  = (const float*)d_in[12];
  const float* ad3  = (const float*)d_in[13];
  const float* b3   = (const float*)d_in[14];
  const float* Wlin = (const float*)d_in[15];
  const float* blin = (const float*)d_in[16];
  float* out = (float*)d_out;

  const int N = in_sizes[2];          // 50000
  const int E = in_sizes[1] / 2;      // 400000
  const int G = 64, F3 = 128, O = 10;

  // workspace layout (floats)
  float* ws = (float*)d_ws;
  const long BUF = (long)N * 512;
  float*    buf0  = ws;
  float*    buf1  = buf0 + BUF;
  float*    buf2  = buf1 + BUF;
  float*    es    = buf2 + BUF;                     // N*4
  float*    ed    = es + (long)N * 4;               // N*4
  unsigned* menc  = (unsigned*)(ed + (long)N * 4);  // N*4
  float*    m     = (float*)menc + (long)N * 4;     // N*4
  float*    den   = m + (long)N * 4;                // N*4
  float*    alpha = den + (long)N * 4;              // (E+N)*4
  float*    ps    = alpha + (long)(E + N) * 4;      // G*F3
  float*    cnt   = ps + (long)G * F3;              // G

  // Layer 1: 128 -> 4x16 concat (64)
  run_gat_layer(x,    N, 128,  16, 1, W1, as1, ad1, b1, ei, E,
                buf0, buf1, es, ed, menc, m, den, alpha, stream);
  // Layer 2: 64 -> 4x64 concat (256)
  run_gat_layer(buf1, N, 64,   64, 1, W2, as2, ad2, b2, ei, E,
                buf2, buf0, es, ed, menc, m, den, alpha, stream);
  // Layer 3: 256 -> 4x128 mean (128)
  run_gat_layer(buf0, N, 256, 128, 0, W3, as3, ad3, b3, ei, E,
                buf1, buf2, es, ed, menc, m, den, alpha, stream);

  // global mean pool + linear head
  hipMemsetAsync(ps,  0, (long)G * F3 * sizeof(float), stream);
  hipMemsetAsync(cnt, 0, G * sizeof(float), stream);
  pool_sum<<<gblk((long)N * F3, 256), 256, 0, stream>>>(buf2, bat, ps, N, F3);
  pool_cnt<<<gblk((long)N, 256), 256, 0, stream>>>(bat, cnt, N);
  final_linear<<<gblk((long)G * O, 64), 64, 0, stream>>>(ps, cnt, Wlin, blin, out, G, F3, O);
}